// MergePanelLabel_84550726189727
// MI455X (gfx1250) — compile-verified
//
#include <hip/hip_runtime.h>
#include <math.h>

// ---------------------------------------------------------------------------
// MergePanelLabel for MI455X (gfx1250, wave32).
// Memory-bound ROI argmax + gather. CDNA5 paths used: async global->LDS
// (global_load_async_to_lds_b128 + s_wait_asynccnt) for the streaming
// class-max pass, wave32 shuffle reductions for the per-cell argmax.
// ---------------------------------------------------------------------------

#define AS1 __attribute__((address_space(1)))
#define AS3 __attribute__((address_space(3)))

typedef int v4i_t __attribute__((ext_vector_type(4)));

#if defined(__has_builtin)
#if __has_builtin(__builtin_amdgcn_global_load_async_to_lds_b128)
#define HAVE_ASYNC_LDS 1
#endif
#if __has_builtin(__builtin_amdgcn_s_wait_asynccnt)
#define HAVE_WAIT_ASYNC 1
#endif
#endif

__device__ __forceinline__ void waitAsync0() {
#if defined(HAVE_WAIT_ASYNC)
  __builtin_amdgcn_s_wait_asynccnt(0);
#elif defined(HAVE_ASYNC_LDS)
  asm volatile("s_wait_asynccnt 0" ::: "memory");
#endif
}

// ---------------------------------------------------------------------------
// Batched contiguous copy (float4 granularity). blockIdx.y = batch index.
// ---------------------------------------------------------------------------
__global__ __launch_bounds__(256) void copy4_kernel(
    float4* __restrict__ dst, const float4* __restrict__ src, int n4,
    long dstStride4, long srcStride4) {
  const long bb = blockIdx.y;
  float4* d = dst + bb * dstStride4;
  const float4* s = src + bb * srcStride4;
  for (int i = blockIdx.x * blockDim.x + threadIdx.x; i < n4;
       i += gridDim.x * blockDim.x)
    d[i] = s[i];
}

// ---------------------------------------------------------------------------
// Per-anchor max over 50 classes. One block = 256 anchors = 51,200 B staged
// through LDS via the async DMA pipe (b128 per lane), then per-thread reduce.
// Grids are exact (all anchor counts divisible by 256).
// ---------------------------------------------------------------------------
__global__ __launch_bounds__(256) void classmax_kernel(
    const float* __restrict__ lc, float* __restrict__ m) {
  const long blockBase = (long)blockIdx.x * 256;
  const int t = threadIdx.x;
  __shared__ __align__(16) float tile[256 * 50];  // 51,200 bytes, 16B aligned
  const float* src = lc + blockBase * 50;

#if defined(HAVE_ASYNC_LDS)
  {
    const char* g = (const char*)src;
    char* lp = (char*)tile;
    for (int ofs = t * 16; ofs < 256 * 200; ofs += 256 * 16) {
      __builtin_amdgcn_global_load_async_to_lds_b128(
          (AS1 v4i_t*)(g + ofs), (AS3 v4i_t*)(lp + ofs), 0, 0);
    }
    waitAsync0();
    __syncthreads();
  }
#else
  {
    float4* lt = (float4*)tile;
    const float4* gt = (const float4*)src;
    for (int i = t; i < 3200; i += 256) lt[i] = gt[i];
    __syncthreads();
  }
#endif

  const float* row = tile + t * 50;
  float mx = row[0];
#pragma unroll
  for (int c = 1; c < 50; ++c) mx = fmaxf(mx, row[c]);
  m[blockBase + t] = mx;
}

// ---------------------------------------------------------------------------
// One wave32 per panel cell: scan all qualifying label-level ROIs in the
// reference's (level, sy, sx, anchor) order; strict-> with (level,k) key
// tie-break reproduces jnp.argmax first-max + earlier-level-wins-ties.
// ---------------------------------------------------------------------------
__device__ __forceinline__ void takeBetter(float& bm, unsigned& bk, float m,
                                           unsigned k) {
  if (m > bm || (m == bm && k < bk)) {
    bm = m;
    bk = k;
  }
}

template <bool USE_M>
__global__ __launch_bounds__(256) void merge_kernel(
    const float* __restrict__ lr0, const float* __restrict__ lr1,
    const float* __restrict__ lr2, const float* __restrict__ lr3,
    const float* __restrict__ lc0, const float* __restrict__ lc1,
    const float* __restrict__ lc2, const float* __restrict__ lc3,
    const float* __restrict__ m0, const float* __restrict__ m1,
    const float* __restrict__ m2, const float* __restrict__ m3,
    float* __restrict__ out_lr, float* __restrict__ out_lc, int p_i, int Hp,
    int baseRow, int nLev) {
  const int gwave = (blockIdx.x * blockDim.x + threadIdx.x) >> 5;
  const int lane = threadIdx.x & 31;
  const int cellsPerB = Hp * Hp;
  const int b = gwave / cellsPerB;
  if (b >= 2) return;
  const int cell = gwave - b * cellsPerB;
  const int yp = cell / Hp;
  const int xp = cell - yp * Hp;

  float bestM = -INFINITY;
  unsigned bestKey = 0xFFFFFFFFu;

#pragma unroll
  for (int l = 0; l < 4; ++l) {
    if (l >= nLev) break;
    const int shift = p_i - l + 2;  // p_layer - l_layer >= 2 always here
    const int s = 1 << shift;
    const int Hl = Hp << shift;
    const int K = s * s * 9;
    const float* mp = (l == 0) ? m0 : (l == 1) ? m1 : (l == 2) ? m2 : m3;
    const float* lcp = (l == 0) ? lc0 : (l == 1) ? lc1 : (l == 2) ? lc2 : lc3;
    for (int k = lane; k < K; k += 32) {
      const int a = k % 9;
      const int cp = k / 9;
      const int sx = cp & (s - 1);
      const int sy = cp >> shift;
      const int y = yp * s + sy;
      const int x = xp * s + sx;
      const long idx = ((((long)b * Hl + y) * Hl) + x) * 9 + a;
      float mv;
      if (USE_M) {
        mv = mp[idx];
      } else {
        const float* c = lcp + idx * 50;
        mv = c[0];
#pragma unroll
        for (int cc = 1; cc < 50; ++cc) mv = fmaxf(mv, c[cc]);
      }
      takeBetter(bestM, bestKey, mv, ((unsigned)l << 16) | (unsigned)k);
    }
  }

  // wave32 butterfly argmax reduction (first-max semantics via key tiebreak)
#pragma unroll
  for (int off = 16; off > 0; off >>= 1) {
    const float om = __shfl_xor(bestM, off, 32);
    const unsigned ok = __shfl_xor(bestKey, off, 32);
    takeBetter(bestM, bestKey, om, ok);
  }

  // decode winner, gather, broadcast across the 9 panel anchors
  const int wl = (int)(bestKey >> 16);
  const int k = (int)(bestKey & 0xFFFFu);
  const int shift = p_i - wl + 2;
  const int s = 1 << shift;
  const int Hl = Hp << shift;
  const int a = k % 9;
  const int cp = k / 9;
  const int sx = cp & (s - 1);
  const int sy = cp >> shift;
  const int y = yp * s + sy;
  const int x = xp * s + sx;
  const long widx = ((((long)b * Hl + y) * Hl) + x) * 9 + a;
  const float* rsrc =
      ((wl == 0) ? lr0 : (wl == 1) ? lr1 : (wl == 2) ? lr2 : lr3) + widx * 4;
  const float* csrc =
      ((wl == 0) ? lc0 : (wl == 1) ? lc1 : (wl == 2) ? lc2 : lc3) + widx * 50;
  const long rowOut = (long)b * 49104 + baseRow + (long)(yp * Hp + xp) * 9;
  float* rd = out_lr + rowOut * 4;
  float* cd = out_lc + rowOut * 50;
  for (int i = lane; i < 9 * 4; i += 32) rd[i] = rsrc[i & 3];
  for (int i = lane; i < 9 * 50; i += 32) cd[i] = csrc[i % 50];
}

// ---------------------------------------------------------------------------
extern "C" void kernel_launch(void* const* d_in, const int* in_sizes, int n_in,
                              void* d_out, int out_size, void* d_ws,
                              size_t ws_size, hipStream_t stream) {
  (void)in_sizes;
  (void)n_in;
  (void)out_size;
  const float* p_r[5];
  const float* p_c[5];
  const float* l_r[4];
  const float* l_c[4];
  for (int i = 0; i < 5; ++i) p_r[i] = (const float*)d_in[i];
  for (int i = 0; i < 5; ++i) p_c[i] = (const float*)d_in[5 + i];
  for (int i = 0; i < 4; ++i) l_r[i] = (const float*)d_in[10 + i];
  for (int i = 0; i < 4; ++i) l_c[i] = (const float*)d_in[14 + i];

  float* out = (float*)d_out;
  const int Hp[5] = {64, 32, 16, 8, 4};
  const int baseRow[5] = {0, 36864, 46080, 48384, 48960};
  const long ROWS = 49104;
  float* out_pr = out;                    // (2, 49104, 4)
  float* out_pc = out + 2 * ROWS * 4;     // (2, 49104, 1)
  float* out_lr = out_pc + 2 * ROWS * 1;  // (2, 49104, 4)
  float* out_lc = out_lr + 2 * ROWS * 4;  // (2, 49104, 50)

  // ---- passthrough copies for p_r / p_c (contiguous per batch+level) ----
  for (int p = 0; p < 5; ++p) {
    const int N = Hp[p] * Hp[p] * 9;  // rows this level (per batch)
    {  // p_r: N rows * 4 floats = N float4s, contiguous both sides
      const int n4 = N;
      dim3 g((unsigned)((n4 + 255) / 256), 2);
      copy4_kernel<<<g, 256, 0, stream>>>((float4*)out_pr + baseRow[p],
                                          (const float4*)p_r[p], n4, ROWS,
                                          (long)N);
    }
    {  // p_c: N floats = N/4 float4s (N, baseRow all divisible by 4)
      const int n4 = N / 4;
      dim3 g((unsigned)((n4 + 255) / 256), 2);
      copy4_kernel<<<g, 256, 0, stream>>>((float4*)(out_pc + baseRow[p]),
                                          (const float4*)p_c[p], n4, ROWS / 4,
                                          (long)(N / 4));
    }
  }

  // ---- pass 1: per-anchor class max into workspace (if it fits) ----
  const long moff[4] = {0, 1179648, 1474560, 1548288};
  const long needF = 1566720;  // total m floats (6.27 MB)
  const int HL[4] = {256, 128, 64, 32};
  const bool useM = (d_ws != nullptr) && (ws_size >= (size_t)needF * 4);
  float* mbase = (float*)d_ws;
  if (useM) {
    for (int l = 0; l < 4; ++l) {
      const long nA = 2L * HL[l] * HL[l] * 9;  // divisible by 256
      classmax_kernel<<<(unsigned)(nA / 256), 256, 0, stream>>>(
          l_c[l], mbase + moff[l]);
    }
  }

  // ---- pass 2: per-panel-cell argmax + gather + broadcast ----
  for (int p = 0; p < 5; ++p) {
    const int nLev = (p + 1 > 4) ? 4 : (p + 1);
    const int cells = 2 * Hp[p] * Hp[p];            // divisible by 8
    const unsigned blocks = (unsigned)(cells / 8);  // 8 waves/block
    if (useM) {
      merge_kernel<true><<<blocks, 256, 0, stream>>>(
          l_r[0], l_r[1], l_r[2], l_r[3], l_c[0], l_c[1], l_c[2], l_c[3],
          mbase + moff[0], mbase + moff[1], mbase + moff[2], mbase + moff[3],
          out_lr, out_lc, p, Hp[p], baseRow[p], nLev);
    } else {
      merge_kernel<false><<<blocks, 256, 0, stream>>>(
          l_r[0], l_r[1], l_r[2], l_r[3], l_c[0], l_c[1], l_c[2], l_c[3],
          nullptr, nullptr, nullptr, nullptr, out_lr, out_lc, p, Hp[p],
          baseRow[p], nLev);
    }
  }
}